// DiceLoss_68908455297535
// MI455X (gfx1250) — compile-verified
//
#include <hip/hip_runtime.h>

// ---------------------------------------------------------------------------
// DiceLoss for MI455X (gfx1250, wave32).
// Memory-bound streaming reduction: 170 MB single-pass -> ~7.3 us floor at
// 23.3 TB/s. Strategy:
//   * 128-bit non-temporal global loads (b128, TH=NT) of pred/target.
//   * sigmoid via v_exp_f32 (exp2) + v_rcp_f32 on the VALU/trans pipe.
//   * Accumulation offloaded to the matrix pipe: V_WMMA_F32_16X16X4_F32 with
//     a ones B-matrix computes D += rowsum(A) — 64 elements absorbed per
//     instruction, order-independent, so coalesced loads feed A directly.
//   * Block-uniform loop bounds (EXEC must be all ones for WMMA); lane
//     masking done by value (multiply by 0/1), never by divergence.
//   * Deterministic 2-stage reduction (no float atomics): per-block partials
//     in d_ws, then one block does a fixed-order tree + the dice formula.
// ---------------------------------------------------------------------------

typedef float v2f __attribute__((ext_vector_type(2)));
typedef float v4f __attribute__((ext_vector_type(4)));
typedef float v8f __attribute__((ext_vector_type(8)));

#define NBLK 1024   // blocks per batch in stage 1
#define TPB  256    // 8 waves of 32
#define WAVES (TPB / 32)

__device__ __forceinline__ float fast_sigmoid(float x) {
  // sigmoid(x) = 1 / (1 + exp(-x)); exp(-x) = exp2(-x * log2(e))
  float e = __builtin_amdgcn_exp2f(x * -1.44269504088896340736f);
  return __builtin_amdgcn_rcpf(1.0f + e);
}

// D += A(16x4) * ones(4x16): every element of each D row accumulates the
// row-sum of A, i.e. one WMMA absorbs 64 new f32 values into the accumulator.
__device__ __forceinline__ v8f wmma_rowsum_acc(v2f a, v8f c) {
  const v2f ones = {1.0f, 1.0f};
  return __builtin_amdgcn_wmma_f32_16x16x4_f32(
      /*neg_a=*/false, a, /*neg_b=*/false, ones,
      /*c_mod=*/(short)0, c, /*reuse_a=*/false, /*reuse_b=*/false);
}

__global__ __launch_bounds__(TPB) void dice_partial_kernel(
    const float* __restrict__ pred, const float* __restrict__ tgt,
    float* __restrict__ partial, long long n_per_batch, int n4, int nblk) {
  const int b = blockIdx.y;
  const float* __restrict__ pb = pred + (long long)b * n_per_batch;
  const float* __restrict__ tb = tgt + (long long)b * n_per_batch;
  const v4f* __restrict__ p4 = (const v4f*)pb;
  const v4f* __restrict__ t4 = (const v4f*)tb;

  v8f accI = {0.f, 0.f, 0.f, 0.f, 0.f, 0.f, 0.f, 0.f};  // sum sig(p)*t
  v8f accP = {0.f, 0.f, 0.f, 0.f, 0.f, 0.f, 0.f, 0.f};  // sum sig(p)
  v8f accT = {0.f, 0.f, 0.f, 0.f, 0.f, 0.f, 0.f, 0.f};  // sum t

  const int stride = nblk * TPB;
  // Loop bound depends only on the block-uniform base index so EXEC stays
  // all-ones for the WMMA instructions; lane validity is applied by value.
  for (int i0 = blockIdx.x * TPB; i0 < n4; i0 += stride) {
    const int i = i0 + (int)threadIdx.x;
    const bool valid = (i < n4);
    const int idx = valid ? i : 0;
    const float m = valid ? 1.0f : 0.0f;

    v4f p = __builtin_nontemporal_load(&p4[idx]);
    v4f t = __builtin_nontemporal_load(&t4[idx]);

    v4f s;
    #pragma unroll
    for (int c = 0; c < 4; ++c) s[c] = fast_sigmoid(p[c]);
    s *= m;
    t *= m;
    v4f st = s * t;

    v2f sa0 = {s[0], s[1]},  sa1 = {s[2], s[3]};
    v2f ta0 = {t[0], t[1]},  ta1 = {t[2], t[3]};
    v2f ia0 = {st[0], st[1]}, ia1 = {st[2], st[3]};

    accP = wmma_rowsum_acc(sa0, accP);
    accT = wmma_rowsum_acc(ta0, accT);
    accI = wmma_rowsum_acc(ia0, accI);
    accP = wmma_rowsum_acc(sa1, accP);
    accT = wmma_rowsum_acc(ta1, accT);
    accI = wmma_rowsum_acc(ia1, accI);
  }

  // Per-lane reduce of the 8 accumulator VGPRs. All columns of D are
  // identical (B = ones), so lanes 0-15 hold sum(rows 0-7) and lanes 16-31
  // hold sum(rows 8-15): one shfl_xor(16) completes the wave total.
  float sI = 0.f, sP = 0.f, sT = 0.f;
  #pragma unroll
  for (int k = 0; k < 8; ++k) { sI += accI[k]; sP += accP[k]; sT += accT[k]; }

  // Scalar tail (n_per_batch % 4), handled once by thread 0 of block (0,b).
  if (blockIdx.x == 0 && threadIdx.x == 0) {
    for (long long i = 4LL * n4; i < n_per_batch; ++i) {
      float p = pb[i], t = tb[i];
      float s = fast_sigmoid(p);
      sP += s; sT += t; sI += s * t;
    }
  }

  sI += __shfl_xor(sI, 16, 32);
  sP += __shfl_xor(sP, 16, 32);
  sT += __shfl_xor(sT, 16, 32);

  __shared__ float red[3][WAVES];
  const int wave = threadIdx.x >> 5;
  if ((threadIdx.x & 31) == 0) {
    red[0][wave] = sI; red[1][wave] = sP; red[2][wave] = sT;
  }
  __syncthreads();
  if (threadIdx.x == 0) {
    float a0 = 0.f, a1 = 0.f, a2 = 0.f;
    #pragma unroll
    for (int w = 0; w < WAVES; ++w) { a0 += red[0][w]; a1 += red[1][w]; a2 += red[2][w]; }
    partial[(b * 3 + 0) * nblk + blockIdx.x] = a0;  // inter
    partial[(b * 3 + 1) * nblk + blockIdx.x] = a1;  // psum
    partial[(b * 3 + 2) * nblk + blockIdx.x] = a2;  // tsum
  }
}

__global__ __launch_bounds__(TPB) void dice_final_kernel(
    const float* __restrict__ partial, const float* __restrict__ weight,
    float* __restrict__ out, int nblk, int nbatch) {
  __shared__ float red[9][TPB];
  const int K = nbatch * 3;  // <= 9
  float acc[9];
  #pragma unroll
  for (int k = 0; k < 9; ++k) acc[k] = 0.f;

  for (int j = threadIdx.x; j < nblk; j += TPB)
    for (int k = 0; k < K; ++k) acc[k] += partial[k * nblk + j];

  for (int k = 0; k < K; ++k) red[k][threadIdx.x] = acc[k];
  __syncthreads();
  for (int s = TPB / 2; s > 0; s >>= 1) {
    if ((int)threadIdx.x < s)
      for (int k = 0; k < K; ++k) red[k][threadIdx.x] += red[k][threadIdx.x + s];
    __syncthreads();
  }

  if (threadIdx.x == 0) {
    const float smooth = 1.0f;
    float total = 0.f;
    for (int b = 0; b < nbatch; ++b) {
      float w = weight[b];
      float inter = red[b * 3 + 0][0] * w;
      float psum  = red[b * 3 + 1][0] * w;
      float tsum  = red[b * 3 + 2][0] * w;
      float dice = (2.0f * inter + smooth) / (psum + tsum + smooth);
      total += 1.0f - dice;
    }
    out[0] = total / (float)nbatch;
  }
}

extern "C" void kernel_launch(void* const* d_in, const int* in_sizes, int n_in,
                              void* d_out, int out_size, void* d_ws, size_t ws_size,
                              hipStream_t stream) {
  const float* pred = (const float*)d_in[0];
  const float* tgt  = (const float*)d_in[1];
  const float* wgt  = (const float*)d_in[2];
  float* out = (float*)d_out;
  float* partial = (float*)d_ws;  // needs 3*3*NBLK floats = 36 KB

  const int B = (n_in > 2 && in_sizes[2] > 0) ? in_sizes[2] : 3;
  const long long n_per_batch = (long long)in_sizes[0] / B;
  const int n4 = (int)(n_per_batch / 4);

  dim3 grid(NBLK, B);
  dice_partial_kernel<<<grid, TPB, 0, stream>>>(pred, tgt, partial,
                                                n_per_batch, n4, NBLK);
  dice_final_kernel<<<1, TPB, 0, stream>>>(partial, wgt, out, NBLK, B);
}